// MultiLabelAdaptiveMarginLoss_73014444032080
// MI455X (gfx1250) — compile-verified
//
#include <hip/hip_runtime.h>

typedef float v2f __attribute__((ext_vector_type(2)));
typedef float v8f __attribute__((ext_vector_type(8)));

#define Bdim 256
#define Cdim 4096
#define Kdim 32
#define THREADS 256

// One block per batch row b.
// LDS: s[j] = input[b,j] - margin[b,j]  (16 KB), c[k] = 1 - s[t_k] (or -1e30 if pad).
// 8 waves: wave = (jQuarter<<1) | kGroup. Each wave owns 16 targets (WMMA M dim)
// and a 1024-wide slice of j, consumed 4 columns per V_WMMA_F32_16X16X4_F32
// with B = ones (row-sum reduction on the matrix pipe).
__global__ __launch_bounds__(THREADS) void mlam_row_kernel(
    const float* __restrict__ input,
    const float* __restrict__ margin,
    const int*   __restrict__ target,
    float*       __restrict__ partial)
{
    __shared__ float s_lds[Cdim];
    __shared__ float c_lds[Kdim];
    __shared__ float bsum;
    __shared__ int   vcnt;

    const int b   = blockIdx.x;
    const int tid = threadIdx.x;

    if (tid == 0) { bsum = 0.0f; vcnt = 0; }

    // Stage s = input - margin into LDS (coalesced float4).
    const float4* inp4 = (const float4*)(input  + (size_t)b * Cdim);
    const float4* mar4 = (const float4*)(margin + (size_t)b * Cdim);
    for (int j4 = tid; j4 < Cdim / 4; j4 += THREADS) {
        float4 x = inp4[j4];
        float4 m = mar4[j4];
        float4 s;
        s.x = x.x - m.x; s.y = x.y - m.y;
        s.z = x.z - m.z; s.w = x.w - m.w;
        ((float4*)s_lds)[j4] = s;
    }
    __syncthreads();

    // c[k] = 1 - s[t_k]; padded labels contribute exactly 0 via c = -1e30.
    if (tid < Kdim) {
        int  t     = target[b * Kdim + tid];
        bool valid = (t > -1);
        float c    = valid ? (1.0f - s_lds[valid ? t : 0]) : -1.0e30f;
        c_lds[tid] = c;
        if (valid) atomicAdd(&vcnt, 1);
    }
    __syncthreads();

    const int lane = tid & 31;
    const int wave = tid >> 5;        // 0..7
    const int kgrp = wave & 1;        // which 16 of the 32 targets
    const int jq   = wave >> 1;       // which 1024-wide quarter of j
    const int hi   = lane >> 4;       // half-wave: K=0,1 vs K=2,3 of the A tile
    const int m    = lane & 15;       // A-matrix row (M)

    const float c    = c_lds[kgrp * 16 + m];
    const v2f   ones = {1.0f, 1.0f};  // B = 4x16 ones -> WMMA does row-sums
    v8f acc = {0.f, 0.f, 0.f, 0.f, 0.f, 0.f, 0.f, 0.f};

    const int jStart = jq * 1024;
    const int jEnd   = jStart + 1024;
    #pragma unroll 4
    for (int j0 = jStart; j0 < jEnd; j0 += 4) {
        const int jA = j0 + (hi << 1);
        float s0 = s_lds[jA];
        float s1 = s_lds[jA + 1];
        v2f a;
        a.x = fmaxf(c + s0, 0.0f);
        a.y = fmaxf(c + s1, 0.0f);
        // D = A(16x4 relu tile) x B(4x16 ones) + C  -> accumulate row sums
        acc = __builtin_amdgcn_wmma_f32_16x16x4_f32(
            /*neg_a=*/false, a, /*neg_b=*/false, ones,
            /*c_mod=*/(short)0, acc, /*reuse_a=*/false, /*reuse_b=*/false);
    }

    // D layout: lanes 0-15 hold identical row-sums for M=0..7 in acc[0..7],
    // lanes 16-31 hold M=8..15. One lane of each half contributes.
    float p = acc[0] + acc[1] + acc[2] + acc[3] +
              acc[4] + acc[5] + acc[6] + acc[7];
    if ((lane & 15) == 0) atomicAdd(&bsum, p);
    __syncthreads();

    if (tid == 0) partial[b] = bsum - (float)vcnt;
}

// Deterministic fixed-order tree reduction of the 256 per-row partials.
__global__ __launch_bounds__(THREADS) void mlam_reduce_kernel(
    const float* __restrict__ partial,
    float*       __restrict__ out)
{
    __shared__ float buf[THREADS];
    const int tid = threadIdx.x;
    buf[tid] = partial[tid];          // Bdim == THREADS == 256
    __syncthreads();
    for (int s = THREADS / 2; s > 0; s >>= 1) {
        if (tid < s) buf[tid] += buf[tid + s];
        __syncthreads();
    }
    if (tid == 0) out[0] = buf[0] * (1.0f / (float)Cdim);
}

extern "C" void kernel_launch(void* const* d_in, const int* in_sizes, int n_in,
                              void* d_out, int out_size, void* d_ws, size_t ws_size,
                              hipStream_t stream) {
    const float* input  = (const float*)d_in[0];
    const float* margin = (const float*)d_in[1];
    const int*   target = (const int*)d_in[2];
    float* partial = (float*)d_ws;     // 256 floats of scratch
    float* out     = (float*)d_out;

    mlam_row_kernel<<<Bdim, THREADS, 0, stream>>>(input, margin, target, partial);
    mlam_reduce_kernel<<<1, THREADS, 0, stream>>>(partial, out);
}